// SpatialAttentionOrig_24257975288473
// MI455X (gfx1250) — compile-verified
//
#include <hip/hip_runtime.h>
#include <math.h>

// Problem constants (from the reference)
#define D1 270
#define KK 32
#define CC 60
#define BB 64
#define TT 4096
#define KL (KK * KK)        // 1024
#define TWO_PI 6.28318530717958647692f

typedef __attribute__((ext_vector_type(2))) float v2f;
typedef __attribute__((ext_vector_type(8))) float v8f;

// ---------------------------------------------------------------------------
// Kernel 1: trig tables  tc[kl*60+c] = cos(2pi(k*x_c + l*y_c)), ts = sin(...)
// Layout [kl][c] so the stage-2 weight GEMM reads coalesced in c.
// ---------------------------------------------------------------------------
__global__ __launch_bounds__(256) void trig_kernel(const float* __restrict__ loc,
                                                   float* __restrict__ tc,
                                                   float* __restrict__ ts) {
    int idx = blockIdx.x * 256 + threadIdx.x;      // 240 blocks * 256 = 61440 exactly
    int kl = idx / CC;
    int c  = idx - kl * CC;
    int k  = kl >> 5;
    int l  = kl & 31;
    float x = loc[c * 2 + 0];
    float y = loc[c * 2 + 1];
    float ph = TWO_PI * ((float)k * x + (float)l * y);
    float s, co;
    sincosf(ph, &s, &co);
    tc[idx] = co;
    ts[idx] = s;
}

// ---------------------------------------------------------------------------
// Kernel 2: a[j,c] = sum_kl z_re[j,kl]*tc[kl,c] + z_im[j,kl]*ts[kl,c];
//           w[j,c] = softmax_c(a[j,c]).  One block per j; thread == c.
// z loads are uniform across the block -> scalar cache path.
// ---------------------------------------------------------------------------
__global__ __launch_bounds__(64) void weights_kernel(const float* __restrict__ z_re,
                                                     const float* __restrict__ z_im,
                                                     const float* __restrict__ tc,
                                                     const float* __restrict__ ts,
                                                     float* __restrict__ wout) {
    int j = blockIdx.x;
    int c = threadIdx.x;                 // 0..63, active c<60
    const float* zr = z_re + (size_t)j * KL;
    const float* zi = z_im + (size_t)j * KL;

    float acc = 0.0f;
    if (c < CC) {
        for (int kl = 0; kl < KL; ++kl) {
            acc = fmaf(zr[kl], tc[kl * CC + c], acc);
            acc = fmaf(zi[kl], ts[kl * CC + c], acc);
        }
    }
    float e = (c < CC) ? expf(acc) : 0.0f;

    __shared__ float red[64];
    red[c] = e;
    __syncthreads();
    for (int s = 32; s > 0; s >>= 1) {
        if (c < s) red[c] += red[c + s];
        __syncthreads();
    }
    float sum = red[0];
    if (c < CC) wout[(size_t)j * CC + c] = e / sum;
}

// ---------------------------------------------------------------------------
// Kernel 3: out[b,j,t] = sum_c w[j,c] * X[b,c,t]   via V_WMMA_F32_16X16X4_F32
//   M = j (17 tiles of 16, rows 270..271 zero-padded in LDS)
//   N = t (16 per wave, 8 waves -> 128-wide t slab per block)
//   K = c (60 = 15 chunks of 4)
// W staged in LDS with row stride 62 floats (248 B: 8B-aligned rows so the
// paired A-fragment reads become ds_load_b64; stride is even -> no 2-way
// bank conflict pattern across 16 consecutive rows).
// X fragments live in registers and are read from HBM exactly once.
// ---------------------------------------------------------------------------
#define WSTRIDE 62
#define JT_TILES 17

__global__ __launch_bounds__(256) void stage2_kernel(const float* __restrict__ X,
                                                     const float* __restrict__ w,
                                                     float* __restrict__ out) {
    __shared__ float Wlds[272 * WSTRIDE];   // 67,456 B < 320 KB

    int tid = threadIdx.x;

    // zero (covers the 2 pad rows), then fill 270x60 from global
    for (int i = tid; i < 272 * WSTRIDE; i += 256) Wlds[i] = 0.0f;
    __syncthreads();
    for (int i = tid; i < D1 * CC; i += 256) {
        int j = i / CC;
        int c = i - j * CC;
        Wlds[j * WSTRIDE + c] = w[i];
    }
    __syncthreads();

    int b    = blockIdx.x >> 5;     // 32 t-slabs per batch
    int tb   = blockIdx.x & 31;
    int wave = tid >> 5;
    int lane = tid & 31;
    int hi   = lane >> 4;           // half-wave select (K pair 0/1 vs 2/3)
    int ln   = lane & 15;           // column within tile
    int t0   = tb * 128 + wave * 16 + ln;

    // B fragments: Bf[cc] holds X[b, cc*4 + 2*hi + {0,1}, t0]
    const float* Xb = X + (size_t)b * CC * TT;
    v2f Bf[15];
#pragma unroll
    for (int cc = 0; cc < 15; ++cc) {
        int c = cc * 4 + hi * 2;
        Bf[cc].x = Xb[(size_t)c * TT + t0];
        Bf[cc].y = Xb[(size_t)(c + 1) * TT + t0];
    }

    for (int jt = 0; jt < JT_TILES; ++jt) {
        int j0 = jt * 16;
        v8f acc = {};
#pragma unroll
        for (int cc = 0; cc < 15; ++cc) {
            int c = cc * 4 + hi * 2;
            const float* wrow = &Wlds[(j0 + ln) * WSTRIDE + c];
            v2f Af;
            Af.x = wrow[0];
            Af.y = wrow[1];
            acc = __builtin_amdgcn_wmma_f32_16x16x4_f32(
                /*neg_a=*/false, Af, /*neg_b=*/false, Bf[cc],
                /*c_mod=*/(short)0, acc, /*reuse_a=*/false, /*reuse_b=*/false);
        }
        // C/D layout: VGPR r -> row (r + hi*8), column ln
        float* ob = out + ((size_t)b * D1 + j0) * TT + t0;
        if (jt < 16) {
#pragma unroll
            for (int r = 0; r < 8; ++r) {
                ob[(size_t)(r + hi * 8) * TT] = acc[r];
            }
        } else {
#pragma unroll
            for (int r = 0; r < 8; ++r) {
                int j = j0 + r + hi * 8;
                if (j < D1) ob[(size_t)(r + hi * 8) * TT] = acc[r];
            }
        }
    }
}

// ---------------------------------------------------------------------------
// launch
// ---------------------------------------------------------------------------
extern "C" void kernel_launch(void* const* d_in, const int* in_sizes, int n_in,
                              void* d_out, int out_size, void* d_ws, size_t ws_size,
                              hipStream_t stream) {
    const float* X    = (const float*)d_in[0];
    const float* z_re = (const float*)d_in[1];
    const float* z_im = (const float*)d_in[2];
    const float* loc  = (const float*)d_in[3];
    float* out = (float*)d_out;

    // workspace: tc(61440) | ts(61440) | w(16200)  => ~556 KB of f32
    float* tc   = (float*)d_ws;
    float* ts   = tc + KL * CC;
    float* wbuf = ts + KL * CC;

    trig_kernel<<<240, 256, 0, stream>>>(loc, tc, ts);
    weights_kernel<<<D1, 64, 0, stream>>>(z_re, z_im, tc, ts, wbuf);
    stage2_kernel<<<BB * (TT / 128), 256, 0, stream>>>(X, wbuf, out);
}